// TripletLoss_22084721836475
// MI455X (gfx1250) — compile-verified
//
#include <hip/hip_runtime.h>
#include <hip/hip_bf16.h>

typedef __attribute__((ext_vector_type(2))) float v2f;
typedef __attribute__((ext_vector_type(8))) float v8f;

#define TL_M        4096                 // rows per subset
#define TL_D        128                  // embedding dim
#define TL_TILES    (TL_M / 16)          // 256 column tiles
#define TL_NSUB     3
#define TL_NEGINF   (-1e30f)

#define WAVES_PER_WG 8
#define WG_ROWS      (WAVES_PER_WG * 16) // 128 rows per workgroup
#define WGS_PER_SUB  (TL_M / WG_ROWS)    // 32
#define N_WGS        (TL_NSUB * WGS_PER_SUB)
#define LDS_STRIDE   132                 // 128 + 4 pad words -> conflict-free b64 reads

__device__ __forceinline__ v8f wmma_f32(v2f a, v2f b, v8f c) {
    // 8 args: (neg_a, A, neg_b, B, c_mod, C, reuse_a, reuse_b)
    return __builtin_amdgcn_wmma_f32_16x16x4_f32(false, a, false, b, (short)0, c,
                                                 false, false);
}

// Cooperatively stage one 16x128 f32 column tile into LDS (padded stride).
__device__ __forceinline__ void fill_tile(float* __restrict__ dst,
                                          const float* __restrict__ src, int tid) {
#pragma unroll
    for (int i = 0; i < 2; ++i) {
        const int f   = tid + i * 256;      // 512 float4s total
        const int row = f >> 5;
        const int c4  = (f & 31) * 4;
        const float4 v = *(const float4*)(src + row * TL_D + c4);
        *(float4*)(dst + row * LDS_STRIDE + c4) = v;
    }
}

// One full sweep over all column tiles. PASS2=false: track row maxima.
// PASS2=true: accumulate thresholded losses.
template <bool PASS2>
__device__ __forceinline__ void
sweep(const float* __restrict__ E, const int* __restrict__ T,
      float (*__restrict__ btile)[16 * LDS_STRIDE],
      const v2f (&a)[32], const int (&trow)[8],
      int rb, int r, int h, int tid,
      float (&maxp)[8], float (&maxn)[8],
      const float (&thrp)[8], const float (&thrn)[8],
      float (&ploss)[8], float (&nloss)[8]) {
    fill_tile(btile[0], E, tid);            // tile 0
    __syncthreads();

    for (int ct = 0; ct < TL_TILES; ++ct) {
        const int buf = ct & 1;
        if (ct + 1 < TL_TILES)              // prefetch next tile into other buffer
            fill_tile(btile[buf ^ 1], E + (size_t)(ct + 1) * 16 * TL_D, tid);

        const int cb   = ct * 16;
        const int tcol = T[cb + r];         // lane = column N in C/D layout
        const int col  = cb + r;

        // 16x16 sim tile: 32 WMMAs, two interleaved accumulator chains
        const float* bl = &btile[buf][r * LDS_STRIDE + 2 * h];
        v8f acc0 = {}, acc1 = {};
#pragma unroll
        for (int kk = 0; kk < 32; kk += 2) {
            v2f b0 = *(const v2f*)(bl + kk * 4);
            v2f b1 = *(const v2f*)(bl + kk * 4 + 4);
            acc0 = wmma_f32(a[kk],     b0, acc0);
            acc1 = wmma_f32(a[kk + 1], b1, acc1);
        }

#pragma unroll
        for (int v = 0; v < 8; ++v) {
            const int   row  = rb + v + 8 * h;
            const bool  same = (trow[v] == tcol);
            const float s    = acc0[v] + acc1[v];
            if (!PASS2) {
                if (same && (col != row)) maxp[v] = fmaxf(maxp[v], s);
                if (!same)                maxn[v] = fmaxf(maxn[v], s);
            } else {
                if (same && (col != row) && (s < thrp[v])) ploss[v] += 1.0f - s;
                if (!same && (s > thrn[v]))                nloss[v] += s;
            }
        }
        __syncthreads();                    // consumption done + next fill visible
    }
}

__global__ __launch_bounds__(WAVES_PER_WG * 32) void
TripletLoss_rows_kernel(const float* __restrict__ emb,
                        const int*   __restrict__ target,
                        float*       __restrict__ partial) {
    __shared__ float btile[2][16 * LDS_STRIDE];
    __shared__ float wpart[WAVES_PER_WG];

    const int wg   = blockIdx.x;
    const int sub  = wg / WGS_PER_SUB;
    const int rblk = wg % WGS_PER_SUB;

    const float* E = emb    + (size_t)sub * TL_M * TL_D;
    const int*   T = target + sub * TL_M;

    const int tid  = threadIdx.x;
    const int wv   = tid >> 5;
    const int lane = tid & 31;
    const int r    = lane & 15;
    const int h    = lane >> 4;
    const int rb   = rblk * WG_ROWS + wv * 16;   // this wave's 16-row tile

    // A fragments: resident in 64 VGPRs across both passes
    v2f a[32];
    const float* arow = E + (size_t)(rb + r) * TL_D + 2 * h;
#pragma unroll
    for (int kk = 0; kk < 32; ++kk)
        a[kk] = *(const v2f*)(arow + kk * 4);

    // Row targets: C/D layout -> vgpr v holds row rb + v + 8*h
    int trow[8];
#pragma unroll
    for (int v = 0; v < 8; ++v)
        trow[v] = T[rb + v + 8 * h];

    float maxp[8], maxn[8], thrp[8], thrn[8], ploss[8], nloss[8];
#pragma unroll
    for (int v = 0; v < 8; ++v) {
        maxp[v] = TL_NEGINF; maxn[v] = TL_NEGINF;
        thrp[v] = 0.0f; thrn[v] = 0.0f; ploss[v] = 0.0f; nloss[v] = 0.0f;
    }

    // ---- pass 1: row maxima over positives / negatives
    sweep<false>(E, T, btile, a, trow, rb, r, h, tid,
                 maxp, maxn, thrp, thrn, ploss, nloss);

    // Reduce max across the 16 lanes of each half (also broadcasts the result)
#pragma unroll
    for (int v = 0; v < 8; ++v) {
#pragma unroll
        for (int off = 1; off < 16; off <<= 1) {
            maxp[v] = fmaxf(maxp[v], __shfl_xor(maxp[v], off, 32));
            maxn[v] = fmaxf(maxn[v], __shfl_xor(maxn[v], off, 32));
        }
    }
#pragma unroll
    for (int v = 0; v < 8; ++v) {
        thrp[v] = maxn[v] + 0.1f;               // pos selected if sim < thrp
        thrn[v] = fmaxf(0.6f, maxp[v]) - 0.1f;  // neg selected if sim > thrn
    }

    // ---- pass 2: thresholded conditional sums (recompute sim)
    sweep<true>(E, T, btile, a, trow, rb, r, h, tid,
                maxp, maxn, thrp, thrn, ploss, nloss);

    // Sum across the 16 lanes of each half
#pragma unroll
    for (int v = 0; v < 8; ++v) {
#pragma unroll
        for (int off = 1; off < 16; off <<= 1) {
            ploss[v] += __shfl_xor(ploss[v], off, 32);
            nloss[v] += __shfl_xor(nloss[v], off, 32);
        }
    }

    float part = 0.0f;
#pragma unroll
    for (int v = 0; v < 8; ++v) {
        const bool has_pos = (maxp[v] > -1e29f);
        if (has_pos) part += ploss[v] + nloss[v];
    }
    part += __shfl_xor(part, 16, 32);           // combine the two row-halves

    if (lane == 0) wpart[wv] = part;
    __syncthreads();
    if (tid == 0) {
        float s = 0.0f;
#pragma unroll
        for (int w = 0; w < WAVES_PER_WG; ++w) s += wpart[w];
        partial[wg] = s;
    }
}

__global__ __launch_bounds__(128) void
TripletLoss_reduce_kernel(const float* __restrict__ partial, float* __restrict__ out) {
    __shared__ float sm[128];
    float s = 0.0f;
    for (int i = threadIdx.x; i < N_WGS; i += 128) s += partial[i];
    sm[threadIdx.x] = s;
    __syncthreads();
#pragma unroll
    for (int st = 64; st > 0; st >>= 1) {
        if ((int)threadIdx.x < st) sm[threadIdx.x] += sm[threadIdx.x + st];
        __syncthreads();
    }
    // mean over rows (4096) then mean over 3 subsets == total / 12288
    if (threadIdx.x == 0) out[0] = sm[0] * (1.0f / (float)(TL_NSUB * TL_M));
}

extern "C" void kernel_launch(void* const* d_in, const int* in_sizes, int n_in,
                              void* d_out, int out_size, void* d_ws, size_t ws_size,
                              hipStream_t stream) {
    const float* emb    = (const float*)d_in[0];
    const int*   target = (const int*)d_in[1];
    float*       ws     = (float*)d_ws;
    float*       out    = (float*)d_out;

    (void)in_sizes; (void)n_in; (void)out_size; (void)ws_size;

    TripletLoss_rows_kernel<<<N_WGS, WAVES_PER_WG * 32, 0, stream>>>(emb, target, ws);
    TripletLoss_reduce_kernel<<<1, 128, 0, stream>>>(ws, out);
}